// Graph_Classification_32220844654960
// MI455X (gfx1250) — compile-verified
//
#include <hip/hip_runtime.h>
#include <hip/hip_bf16.h>

#define NNODES 100000
#define NEDGES 1600000
#define FIN    64
#define HID    128
#define OUTF   10
#define NLAYERS 3

typedef __attribute__((ext_vector_type(2))) float v2f;
typedef __attribute__((ext_vector_type(8))) float v8f;

// ---------------------------------------------------------------- utilities

__global__ void zero_f32_kernel(float* __restrict__ p, int n) {
    int i = blockIdx.x * blockDim.x + threadIdx.x;
    int stride = gridDim.x * blockDim.x;
    for (; i < n; i += stride) p[i] = 0.0f;
}

__global__ void degree_kernel(const int* __restrict__ dst, float* __restrict__ deg, int e) {
    int i = blockIdx.x * blockDim.x + threadIdx.x;
    if (i < e) atomicAdd(&deg[dst[i]], 1.0f);
}

__global__ void invdeg_kernel(float* __restrict__ deg, int n) {
    int i = blockIdx.x * blockDim.x + threadIdx.x;
    if (i < n) deg[i] = 1.0f / fmaxf(deg[i], 1.0f);
}

// one wave32 per edge; each lane moves one float4 (128 feats = 32 float4)
__global__ void scatter_kernel(const float* __restrict__ h, const int* __restrict__ src,
                               const int* __restrict__ dst, float* __restrict__ agg, int e) {
    int t = blockIdx.x * blockDim.x + threadIdx.x;
    int w = t >> 5, lane = t & 31;
    if (w >= e) return;
    size_t s = (size_t)src[w] * HID;
    size_t d = (size_t)dst[w] * HID;
    const float4 v = *(const float4*)(h + s + lane * 4);
    float* b = agg + d + lane * 4;
    atomicAdd(b + 0, v.x);
    atomicAdd(b + 1, v.y);
    atomicAdd(b + 2, v.z);
    atomicAdd(b + 3, v.w);
}

// ---------------------------------------------------------------- GEMM (fp32 WMMA)
// Block: 256 threads = 8 wave32; 16 rows x 128 cols per block; each wave one 16x16 tile.
// A tile staged in LDS with stride K+4 (all 32 lanes hit distinct banks).

template <int K, bool RELU>
__global__ __launch_bounds__(256) void gemm_bias_kernel(const float* __restrict__ A,
                                                        const float* __restrict__ W,
                                                        const float* __restrict__ bias,
                                                        float* __restrict__ Y) {
    constexpr int LDA = K + 4;
    __shared__ float lds_a[16 * LDA];
    const int t = threadIdx.x;
    const int row0 = blockIdx.x * 16;

    constexpr int NF4 = 16 * K / 4;  // float4 count in tile
    constexpr int F4R = K / 4;       // float4 per row
    for (int i = t; i < NF4; i += 256) {
        int row = i / F4R, c4 = i % F4R;
        const float4 v = *(const float4*)(A + (size_t)(row0 + row) * K + c4 * 4);
        float* p = &lds_a[row * LDA + c4 * 4];
        p[0] = v.x; p[1] = v.y; p[2] = v.z; p[3] = v.w;
    }
    __syncthreads();

    const int wave = t >> 5, lane = t & 31;
    const int half = lane >> 4, n = lane & 15;
    const int col0 = wave * 16;

    v8f c = {};
    #pragma unroll 4
    for (int k = 0; k < K; k += 4) {
        v2f a, b;
        a[0] = lds_a[n * LDA + k + 2 * half];
        a[1] = lds_a[n * LDA + k + 2 * half + 1];
        b[0] = W[(size_t)(k + 2 * half) * HID + col0 + n];
        b[1] = W[(size_t)(k + 2 * half + 1) * HID + col0 + n];
        c = __builtin_amdgcn_wmma_f32_16x16x4_f32(false, a, false, b, (short)0, c, false, false);
    }

    const float bv = bias[col0 + n];
    #pragma unroll
    for (int r = 0; r < 8; r++) {
        float v = c[r] + bv;
        if (RELU) v = fmaxf(v, 0.0f);
        Y[(size_t)(row0 + r + 8 * half) * HID + col0 + n] = v;
    }
}

// ---------------------------------------------------------------- fused SAGE layer
// h_out = LayerNorm(ReLU(h@Ws + (agg*inv)@Wn + b)) ; 16 rows x 128 cols per block.

__global__ __launch_bounds__(256) void sage_kernel(const float* __restrict__ hin,
                                                   const float* __restrict__ agg,
                                                   const float* __restrict__ inv,
                                                   const float* __restrict__ Ws,
                                                   const float* __restrict__ Wn,
                                                   const float* __restrict__ bias,
                                                   const float* __restrict__ gam,
                                                   const float* __restrict__ bet,
                                                   float* __restrict__ hout) {
    constexpr int K = HID, LDA = K + 4;
    __shared__ float lds_a[16 * LDA];
    __shared__ float s_mu[16], s_rs[16];
    const int t = threadIdx.x;
    const int row0 = blockIdx.x * 16;
    const int wave = t >> 5, lane = t & 31;
    const int half = lane >> 4, n = lane & 15;
    const int col0 = wave * 16;

    // phase 1: self term
    for (int i = t; i < 512; i += 256) {
        int row = i >> 5, c4 = i & 31;
        const float4 v = *(const float4*)(hin + (size_t)(row0 + row) * HID + c4 * 4);
        float* p = &lds_a[row * LDA + c4 * 4];
        p[0] = v.x; p[1] = v.y; p[2] = v.z; p[3] = v.w;
    }
    __syncthreads();

    v8f c = {};
    #pragma unroll 4
    for (int k = 0; k < K; k += 4) {
        v2f a, b;
        a[0] = lds_a[n * LDA + k + 2 * half];
        a[1] = lds_a[n * LDA + k + 2 * half + 1];
        b[0] = Ws[(size_t)(k + 2 * half) * HID + col0 + n];
        b[1] = Ws[(size_t)(k + 2 * half + 1) * HID + col0 + n];
        c = __builtin_amdgcn_wmma_f32_16x16x4_f32(false, a, false, b, (short)0, c, false, false);
    }
    __syncthreads();

    // phase 2: neighbor term (mean = agg * inv_deg, folded into A-tile load)
    for (int i = t; i < 512; i += 256) {
        int row = i >> 5, c4 = i & 31;
        const float sc = inv[row0 + row];
        const float4 v = *(const float4*)(agg + (size_t)(row0 + row) * HID + c4 * 4);
        float* p = &lds_a[row * LDA + c4 * 4];
        p[0] = v.x * sc; p[1] = v.y * sc; p[2] = v.z * sc; p[3] = v.w * sc;
    }
    __syncthreads();

    #pragma unroll 4
    for (int k = 0; k < K; k += 4) {
        v2f a, b;
        a[0] = lds_a[n * LDA + k + 2 * half];
        a[1] = lds_a[n * LDA + k + 2 * half + 1];
        b[0] = Wn[(size_t)(k + 2 * half) * HID + col0 + n];
        b[1] = Wn[(size_t)(k + 2 * half + 1) * HID + col0 + n];
        c = __builtin_amdgcn_wmma_f32_16x16x4_f32(false, a, false, b, (short)0, c, false, false);
    }

    // bias + ReLU, stash the tile back into LDS for the row-wise LayerNorm
    __syncthreads();
    const float bv = bias[col0 + n];
    #pragma unroll
    for (int r = 0; r < 8; r++) {
        float v = fmaxf(c[r] + bv, 0.0f);
        lds_a[(r + 8 * half) * LDA + col0 + n] = v;
    }
    __syncthreads();

    if (t < 16) {
        float s = 0.0f, ss = 0.0f;
        for (int j = 0; j < HID; j++) {
            float v = lds_a[t * LDA + j];
            s += v; ss += v * v;
        }
        float mu  = s * (1.0f / HID);
        float var = fmaxf(ss * (1.0f / HID) - mu * mu, 0.0f);
        s_mu[t] = mu;
        s_rs[t] = rsqrtf(var + 1e-5f);
    }
    __syncthreads();

    #pragma unroll
    for (int i = 0; i < 8; i++) {
        int idx = t + i * 256;
        int row = idx >> 7, col = idx & 127;
        float v = lds_a[row * LDA + col];
        hout[(size_t)(row0 + row) * HID + col] =
            gam[col] * (v - s_mu[row]) * s_rs[row] + bet[col];
    }
}

// ---------------------------------------------------------------- reductions / head

__global__ __launch_bounds__(256) void colsum_kernel(const float* __restrict__ h,
                                                     float* __restrict__ gsum) {
    __shared__ float red[256];
    const int t = threadIdx.x;
    const int row0 = blockIdx.x * 16;
    const int col = t & 127, rp = t >> 7;
    float s = 0.0f;
    for (int r = rp; r < 16; r += 2)
        s += h[(size_t)(row0 + r) * HID + col];
    red[t] = s;
    __syncthreads();
    if (t < 128) atomicAdd(&gsum[col], red[t] + red[t + 128]);
}

__global__ void head_kernel(const float* __restrict__ gsum, const float* __restrict__ Wout,
                            const float* __restrict__ bout, float* __restrict__ out) {
    int t = threadIdx.x;
    if (t >= OUTF) return;
    float acc = bout[t];
    const float scale = 1.0f / (float)NNODES;
    for (int k = 0; k < HID; k++)
        acc += (gsum[k] * scale) * Wout[k * OUTF + t];
    out[t] = acc;
}

// ---------------------------------------------------------------- launcher

extern "C" void kernel_launch(void* const* d_in, const int* in_sizes, int n_in,
                              void* d_out, int out_size, void* d_ws, size_t ws_size,
                              hipStream_t stream) {
    (void)in_sizes; (void)n_in; (void)out_size; (void)ws_size;

    const float* nodes  = (const float*)d_in[0];
    const int*   src    = (const int*)  d_in[1];
    const int*   dst    = (const int*)  d_in[2];
    const float* W_in1  = (const float*)d_in[3];
    const float* b_in1  = (const float*)d_in[4];
    const float* W_in2  = (const float*)d_in[5];
    const float* b_in2  = (const float*)d_in[6];
    const float* W_self = (const float*)d_in[7];
    const float* W_neigh= (const float*)d_in[8];
    const float* b_sage = (const float*)d_in[9];
    const float* ln_g   = (const float*)d_in[10];
    const float* ln_b   = (const float*)d_in[11];
    const float* W_out  = (const float*)d_in[12];
    const float* b_out  = (const float*)d_in[13];
    float* out = (float*)d_out;

    const size_t NH = (size_t)NNODES * HID;
    char* ws = (char*)d_ws;
    float* hA   = (float*)(ws);
    float* hB   = (float*)(ws + NH * sizeof(float));
    float* agg  = (float*)(ws + 2 * NH * sizeof(float));
    float* inv  = (float*)(ws + 3 * NH * sizeof(float));
    float* gsum = (float*)(ws + 3 * NH * sizeof(float) + (size_t)NNODES * sizeof(float));

    const int rowBlocks = NNODES / 16;                  // 6250 (N % 16 == 0)
    const dim3 blk(256);

    // degree -> inv_deg
    zero_f32_kernel<<<dim3(512), blk, 0, stream>>>(inv, NNODES);
    degree_kernel<<<dim3((NEDGES + 255) / 256), blk, 0, stream>>>(dst, inv, NEDGES);
    invdeg_kernel<<<dim3((NNODES + 255) / 256), blk, 0, stream>>>(inv, NNODES);

    // input MLP:  hB = ReLU(nodes @ W_in1 + b_in1) ; hA = hB @ W_in2 + b_in2
    gemm_bias_kernel<FIN, true ><<<dim3(rowBlocks), blk, 0, stream>>>(nodes, W_in1, b_in1, hB);
    gemm_bias_kernel<HID, false><<<dim3(rowBlocks), blk, 0, stream>>>(hB, W_in2, b_in2, hA);

    // SAGE layers
    float* cur = hA;
    float* nxt = hB;
    for (int l = 0; l < NLAYERS; l++) {
        zero_f32_kernel<<<dim3(4096), blk, 0, stream>>>(agg, (int)NH);
        scatter_kernel<<<dim3(NEDGES / 8), blk, 0, stream>>>(cur, src, dst, agg, NEDGES);
        sage_kernel<<<dim3(rowBlocks), blk, 0, stream>>>(
            cur, agg, inv,
            W_self  + (size_t)l * HID * HID,
            W_neigh + (size_t)l * HID * HID,
            b_sage  + (size_t)l * HID,
            ln_g    + (size_t)l * HID,
            ln_b    + (size_t)l * HID,
            nxt);
        float* tmp = cur; cur = nxt; nxt = tmp;
    }

    // mean over nodes + output head
    zero_f32_kernel<<<dim3(1), blk, 0, stream>>>(gsum, HID);
    colsum_kernel<<<dim3(rowBlocks), blk, 0, stream>>>(cur, gsum);
    head_kernel<<<dim3(1), dim3(32), 0, stream>>>(gsum, W_out, b_out, out);
}